// OFFM_85839216377875
// MI455X (gfx1250) — compile-verified
//
#include <hip/hip_runtime.h>

typedef float v2f __attribute__((ext_vector_type(2)));
typedef float v8f __attribute__((ext_vector_type(8)));

#define CIN    256   // input channels (all levels)
#define HIDDEN 64    // conv1 output channels
#define CIN2   96    // conv2 input channels (HIDDEN + HIDDEN/2)

// ---------------------------------------------------------------------------
// Tiny weight transpose:  src[rows][cols] -> dst[cols][rows]
// Makes the K (input-channel) dimension contiguous so WMMA A-tiles are
// single global_load_b64's in the main kernel.
// ---------------------------------------------------------------------------
__global__ void wtranspose(const float* __restrict__ src, float* __restrict__ dst,
                           int rows, int cols) {
    int idx = blockIdx.x * blockDim.x + threadIdx.x;
    if (idx < rows * cols) {
        int r = idx / cols, c = idx - r * cols;
        dst[c * rows + r] = src[idx];
    }
}

// ---------------------------------------------------------------------------
// Fused level kernel: O = W2^T * concat(Y[0:64], relu(Y[32:64])) + b2,
//                     Y = W1^T * X + b1
// Layouts:  x,out: [4][256][HW],  w1t: [64][256],  w2t: [256][96]
// Block = 256 threads = 8 waves; each wave owns 16 consecutive pixels and
// computes all 256 output channels for them via V_WMMA_F32_16X16X4_F32.
// HW is a template parameter so all channel strides fold to immediates.
// WMMA operand mapping (per ISA 7.12.2):
//   A (16x4):  lane l holds M = (l&15), K = k0 + 2*(l>>4) + {0,1}
//   B (4x16):  lane l holds N = (l&15), K = k0 + 2*(l>>4) + {0,1}
//   C/D:       VGPR r, lane l -> row = r + 8*(l>>4), col = (l&15)
// ---------------------------------------------------------------------------
template <int HW>
__global__ __launch_bounds__(256) void fused_level(
    const float* __restrict__ x,
    const float* __restrict__ w1t, const float* __restrict__ b1,
    const float* __restrict__ w2t, const float* __restrict__ b2,
    float* __restrict__ out)
{
    __shared__ float ylds[8][HIDDEN * 17];   // pad-17 stride: conflict-free

    const int lane = threadIdx.x & 31;
    const int wave = threadIdx.x >> 5;
    const int col  = lane & 15;
    const int half = lane >> 4;

    const int pixBase = blockIdx.x * 128 + wave * 16;   // HW % 128 == 0 always
    const int b  = pixBase / HW;                        // power of two: shift
    const int hw = pixBase - b * HW;

    const float* xcol = x + (size_t)b * CIN * HW + hw + col;
    float* my = &ylds[wave][0];

    // ---------------- stage 1: Y(64x16) = W1^T * X ----------------
    v8f acc1[4] = {};
#pragma unroll 2
    for (int k0 = 0; k0 < CIN; k0 += 4) {
        const int kr = k0 + 2 * half;
        v2f bm;
        bm.x = xcol[(size_t)kr * HW];
        bm.y = xcol[(size_t)(kr + 1) * HW];
        if (k0 + 12 < CIN)                       // uniform guard: stay in-buffer
            __builtin_prefetch(&xcol[(size_t)(kr + 12) * HW], 0, 3);
#pragma unroll
        for (int t = 0; t < 4; ++t) {
            v2f am = *(const v2f*)(w1t + (t * 16 + col) * CIN + kr);
            acc1[t] = __builtin_amdgcn_wmma_f32_16x16x4_f32(
                false, am, false, bm, (short)0, acc1[t], false, false);
        }
    }

    // Y + b1 -> LDS (row-major, stride 17)
#pragma unroll
    for (int t = 0; t < 4; ++t) {
#pragma unroll
        for (int r = 0; r < 8; ++r) {
            const int row = t * 16 + r + 8 * half;
            my[row * 17 + col] = acc1[t][r] + b1[row];
        }
    }
    __syncthreads();

    // ------- stage 2: O(256x16) = W2^T * [Y ; relu(Y[32:64])] -------
    float* o = out + (size_t)b * CIN * HW + hw + col;
    for (int g = 0; g < 2; ++g) {                 // 2 groups of 8 M-tiles
        v8f acc2[8] = {};
        for (int ks = 0; ks < 24; ++ks) {         // K = 96
            const int kr   = ks * 4 + 2 * half;
            const int krow = (ks >= 16) ? (kr - 32) : kr;  // relu rows reuse Y[32:64]
            float z0 = my[krow * 17 + col];
            float z1 = my[krow * 17 + 17 + col];
            if (ks >= 16) { z0 = fmaxf(z0, 0.0f); z1 = fmaxf(z1, 0.0f); }
            v2f bm; bm.x = z0; bm.y = z1;
#pragma unroll
            for (int t = 0; t < 8; ++t) {
                v2f am = *(const v2f*)(w2t + ((g * 8 + t) * 16 + col) * CIN2 + kr);
                acc2[t] = __builtin_amdgcn_wmma_f32_16x16x4_f32(
                    false, am, false, bm, (short)0, acc2[t], false, false);
            }
        }
#pragma unroll
        for (int t = 0; t < 8; ++t) {
#pragma unroll
            for (int r = 0; r < 8; ++r) {
                const int row = (g * 8 + t) * 16 + r + 8 * half;
                o[(size_t)row * HW] = acc2[t][r] + b2[row];
            }
        }
    }
}

// ---------------------------------------------------------------------------
// d_in order: x0,x1,x2, then per level i: W1_i, b1_i, W2_i, b2_i
// d_out: o0 | o1 | o2 concatenated (fp32)
// ---------------------------------------------------------------------------
extern "C" void kernel_launch(void* const* d_in, const int* in_sizes, int n_in,
                              void* d_out, int out_size, void* d_ws, size_t ws_size,
                              hipStream_t stream) {
    (void)in_sizes; (void)n_in; (void)out_size; (void)ws_size;
    static const int HWs[3] = {128 * 128, 64 * 64, 32 * 32};

    float* ws  = (float*)d_ws;
    float* out = (float*)d_out;

    float* w1t[3];
    float* w2t[3];
    size_t wsOff = 0;
    for (int i = 0; i < 3; ++i) {
        const float* W1 = (const float*)d_in[3 + 4 * i];
        const float* W2 = (const float*)d_in[5 + 4 * i];
        w1t[i] = ws + wsOff; wsOff += (size_t)HIDDEN * CIN;  // 16384 floats
        w2t[i] = ws + wsOff; wsOff += (size_t)CIN * CIN2;    // 24576 floats
        wtranspose<<<(CIN * HIDDEN + 255) / 256, 256, 0, stream>>>(W1, w1t[i], CIN, HIDDEN);
        wtranspose<<<(CIN2 * CIN + 255) / 256, 256, 0, stream>>>(W2, w2t[i], CIN2, CIN);
    }

    size_t outOff = 0;
    for (int i = 0; i < 3; ++i) {
        const float* x  = (const float*)d_in[i];
        const float* b1 = (const float*)d_in[4 + 4 * i];
        const float* b2 = (const float*)d_in[6 + 4 * i];
        const int HW = HWs[i];
        const int nblocks = (4 * HW) / 128;   // 16 px/wave * 8 waves
        switch (i) {
            case 0: fused_level<128 * 128><<<nblocks, 256, 0, stream>>>(
                        x, w1t[i], b1, w2t[i], b2, out + outOff); break;
            case 1: fused_level<64 * 64><<<nblocks, 256, 0, stream>>>(
                        x, w1t[i], b1, w2t[i], b2, out + outOff); break;
            default: fused_level<32 * 32><<<nblocks, 256, 0, stream>>>(
                        x, w1t[i], b1, w2t[i], b2, out + outOff); break;
        }
        outOff += (size_t)4 * CIN * HW;
    }
}